// DK_50414326120800
// MI455X (gfx1250) — compile-verified
//
#include <hip/hip_runtime.h>

// ---------------- problem constants ----------------
#define BB   16
#define CC   256     // C == CI == 256
#define HWP  4096    // H*W
#define WW   64
#define EPSV 1e-5f

typedef float v2f __attribute__((ext_vector_type(2)));
typedef float v8f __attribute__((ext_vector_type(8)));

// D = A(16x4) * B(4x16) + C, all f32, wave32 WMMA
__device__ __forceinline__ v8f wmma4(v2f a, v2f b, v8f c) {
    return __builtin_amdgcn_wmma_f32_16x16x4_f32(false, a, false, b, (short)0, c,
                                                 false, false);
}

// ---- gfx1250 async global->LDS path (guarded; falls back to float4 copy) ----
#if defined(__AMDGCN__) && __has_builtin(__builtin_amdgcn_global_load_async_to_lds_b128)
#define HAVE_ASYNC_LDS 1
typedef int v4i __attribute__((vector_size(16)));
typedef __attribute__((address_space(1))) v4i gv4i;   // global (AS1) int4
typedef __attribute__((address_space(3))) v4i lv4i;   // LDS    (AS3) int4
__device__ __forceinline__ void async_copy16(const float* g, float* l) {
    __builtin_amdgcn_global_load_async_to_lds_b128((gv4i*)g, (lv4i*)l, 0, 0);
}
__device__ __forceinline__ void async_wait0() {
#if __has_builtin(__builtin_amdgcn_s_wait_asynccnt)
    __builtin_amdgcn_s_wait_asynccnt(0);
#else
    asm volatile("s_wait_asynccnt 0" ::: "memory");
#endif
}
#endif

// ---------------------------------------------------------------------------
// 16x16 adaptive average pool:  x[plane][64][64] -> pooled[plane][16]
// ---------------------------------------------------------------------------
__global__ __launch_bounds__(256)
void pool16(const float* __restrict__ x, float* __restrict__ pooled) {
    __shared__ float part[256];
    const int plane = blockIdx.x;
    const float* xp = x + (size_t)plane * HWP;
    const int t  = threadIdx.x;
    const int s  = t >> 4;        // pool cell 0..15
    const int iy = t & 15;        // row within cell
    const int by = s >> 2, bx = s & 3;
    const float* row = xp + (by * 16 + iy) * WW + bx * 16;
    float sum = 0.f;
#pragma unroll
    for (int i = 0; i < 4; ++i) {
        float4 v = *reinterpret_cast<const float4*>(row + i * 4);
        sum += v.x + v.y + v.z + v.w;
    }
    part[t] = sum;
    __syncthreads();
    if (t < 16) {
        float acc = 0.f;
#pragma unroll
        for (int i = 0; i < 16; ++i) acc += part[t * 16 + i];
        pooled[(size_t)plane * 16 + t] = acc * (1.0f / 256.0f);
    }
}

// ---------------------------------------------------------------------------
// y[b][o][p] = sum_k w[o][k] * x[b][k][p] + bias[o]   (M=256, K=256, N tile NT)
// 512 threads = 16 waves; wave w owns output-channel strip [16w, 16w+16) and
// computes a 16xNT tile (NT/16 accumulators share each A fetch).
// x-tile staged in LDS with padded row stride (bank-conflict-free for NT=32).
// Optionally accumulates per-channel sum / sum-of-squares for BatchNorm.
// ---------------------------------------------------------------------------
template <int NT>
__global__ __launch_bounds__(512)
void gemm256(const float* __restrict__ x, const float* __restrict__ w,
             const float* __restrict__ bias, float* __restrict__ y,
             float* __restrict__ stats, int Np) {
    constexpr int NTILES = NT / 16;
    constexpr int LST = (NT == 32) ? 40 : 16;   // padded LDS row stride (floats)
    __shared__ float lx[256 * LST];

    const int b   = blockIdx.z;
    const int p0  = blockIdx.x * NT;
    const int tid = threadIdx.x;
    const int wave  = tid >> 5;
    const int lane  = tid & 31;
    const int hlane = lane & 15;
    const int hi    = lane >> 4;          // 0: K pair {k,k+1}, 1: {k+2,k+3}

    // cooperative tile load: 256 rows x NT pixels, 16B chunks
    const float* xb = x + (size_t)b * 256 * Np + p0;
    constexpr int CHUNKS = 256 * NT / 4;
    for (int i = tid; i < CHUNKS; i += 512) {
        const int ci = i / (NT / 4), q = i % (NT / 4);
        const float* src = xb + (size_t)ci * Np + q * 4;
        float* dst = &lx[ci * LST + q * 4];
#ifdef HAVE_ASYNC_LDS
        async_copy16(src, dst);               // global_load_async_to_lds_b128
#else
        *reinterpret_cast<float4*>(dst) = *reinterpret_cast<const float4*>(src);
#endif
    }
#ifdef HAVE_ASYNC_LDS
    async_wait0();                            // s_wait_asynccnt 0
#endif
    __syncthreads();

    const int o0   = wave * 16;
    const int orow = o0 + hlane;
    const float* wrow = w + (size_t)orow * 256;
    __builtin_prefetch(wrow, 0, 0);           // global_prefetch_b8 (L2-hot weights)

    v8f acc[NTILES];
#pragma unroll
    for (int nt = 0; nt < NTILES; ++nt)
        acc[nt] = (v8f){0.f, 0.f, 0.f, 0.f, 0.f, 0.f, 0.f, 0.f};

#pragma unroll 4
    for (int k = 0; k < 256; k += 4) {
        const int ks = k + hi * 2;
        const float2 wv = *reinterpret_cast<const float2*>(wrow + ks);
        v2f a;  a.x = wv.x;  a.y = wv.y;      // A: rows=o, K pair
#pragma unroll
        for (int nt = 0; nt < NTILES; ++nt) { // A reused across N tiles
            v2f bm;
            bm.x = lx[ks * LST + nt * 16 + hlane];
            bm.y = lx[(ks + 1) * LST + nt * 16 + hlane];
            acc[nt] = wmma4(a, bm, acc[nt]);
        }
    }

    // epilogue: bias, store, BN statistics
    float* yb = y + (size_t)b * 256 * Np + p0 + hlane;
#pragma unroll
    for (int j = 0; j < 8; ++j) {
        const int r = o0 + j + hi * 8;        // output channel
        const float bv = bias[r];
        float s = 0.f, q = 0.f;
#pragma unroll
        for (int nt = 0; nt < NTILES; ++nt) {
            float v = acc[nt][j] + bv;
            yb[(size_t)r * Np + nt * 16] = v;
            s += v;
            q += v * v;
        }
        if (stats) {
#pragma unroll
            for (int m = 1; m < 16; m <<= 1) {
                s += __shfl_xor(s, m, 16);
                q += __shfl_xor(q, m, 16);
            }
            if (hlane == 0) {
                atomicAdd(&stats[r], s);
                atomicAdd(&stats[256 + r], q);
            }
        }
    }
}

// ---------------------------------------------------------------------------
// BatchNorm finalize: stats -> per-channel (scale a, shift b)
// ---------------------------------------------------------------------------
__global__ __launch_bounds__(256)
void bnfin(const float* __restrict__ stats, const float* __restrict__ gamma,
           const float* __restrict__ beta, float* __restrict__ ab, float inv_n) {
    const int c = threadIdx.x;
    const float mean = stats[c] * inv_n;
    const float var  = stats[256 + c] * inv_n - mean * mean;
    const float a    = gamma[c] * rsqrtf(var + EPSV);
    ab[2 * c]     = a;
    ab[2 * c + 1] = beta[c] - mean * a;
}

// ---------------------------------------------------------------------------
// Per-(b,c) 4x4 dynamic cross-correlation with BN+ReLU applied on load.
// padding: top/left = 2, bottom/right = 1  ->  in(h-2+i, w-2+j)
// ---------------------------------------------------------------------------
__global__ __launch_bounds__(256)
void dynconv(const float* __restrict__ y, const float* __restrict__ ab,
             const float* __restrict__ ker, float* __restrict__ g) {
    __shared__ float tile[64 * 65];
    __shared__ float kk[16];
    const int plane = blockIdx.x;             // b*256 + c
    const int c = plane & 255;
    const float a = ab[2 * c], bs = ab[2 * c + 1];
    const float* yp = y + (size_t)plane * HWP;
    const int t = threadIdx.x;
    if (t < 16) kk[t] = ker[(size_t)plane * 16 + t];
#pragma unroll
    for (int i = 0; i < 16; ++i) {
        const int idx = i * 256 + t;
        const int r = idx >> 6, w = idx & 63;
        float v = yp[idx];
        tile[r * 65 + w] = fmaxf(fmaf(a, v, bs), 0.f);   // BN + ReLU fused
    }
    __syncthreads();

    const int w = t & 63;
    const int hbase = (t >> 6) * 16;
    float* gp = g + (size_t)plane * HWP;
    for (int r = 0; r < 16; ++r) {
        const int h = hbase + r;
        float acc = 0.f;
#pragma unroll
        for (int i = 0; i < 4; ++i) {
            const int hy = h - 2 + i;
            if (hy < 0 || hy > 63) continue;
#pragma unroll
            for (int j = 0; j < 4; ++j) {
                const int wx = w - 2 + j;
                if (wx >= 0 && wx <= 63)
                    acc = fmaf(tile[hy * 65 + wx], kk[i * 4 + j], acc);
            }
        }
        gp[h * 64 + w] = acc;
    }
}

// ---------------------------------------------------------------------------
// Elementwise BN + ReLU, in place, float4 wide. channel = (i >> 12) & 255.
// ---------------------------------------------------------------------------
__global__ __launch_bounds__(256)
void bnrelu_ew(float* __restrict__ z, const float* __restrict__ ab) {
    const size_t i = ((size_t)blockIdx.x * 256 + threadIdx.x) * 4;
    const int c = (int)((i >> 12) & 255);
    const float a = ab[2 * c], bs = ab[2 * c + 1];
    float4 v = *reinterpret_cast<float4*>(z + i);
    v.x = fmaxf(fmaf(a, v.x, bs), 0.f);
    v.y = fmaxf(fmaf(a, v.y, bs), 0.f);
    v.z = fmaxf(fmaf(a, v.z, bs), 0.f);
    v.w = fmaxf(fmaf(a, v.w, bs), 0.f);
    *reinterpret_cast<float4*>(z + i) = v;
}

// ---------------------------------------------------------------------------
extern "C" void kernel_launch(void* const* d_in, const int* in_sizes, int n_in,
                              void* d_out, int out_size, void* d_ws, size_t ws_size,
                              hipStream_t stream) {
    (void)in_sizes; (void)n_in; (void)out_size; (void)ws_size;

    const float* xf   = (const float*)d_in[0];
    const float* xe   = (const float*)d_in[1];
    const float* w_kf = (const float*)d_in[2];
    const float* b_kf = (const float*)d_in[3];
    const float* w_ke = (const float*)d_in[4];
    const float* b_ke = (const float*)d_in[5];
    const float* w_rf = (const float*)d_in[6];
    const float* b_rf = (const float*)d_in[7];
    const float* g_rf = (const float*)d_in[8];
    const float* e_rf = (const float*)d_in[9];
    const float* w_re = (const float*)d_in[10];
    const float* b_re = (const float*)d_in[11];
    const float* g_re = (const float*)d_in[12];
    const float* e_re = (const float*)d_in[13];
    const float* w_bf = (const float*)d_in[14];
    const float* b_bf = (const float*)d_in[15];
    const float* g_bf = (const float*)d_in[16];
    const float* e_bf = (const float*)d_in[17];
    const float* w_be = (const float*)d_in[18];
    const float* b_be = (const float*)d_in[19];
    const float* g_be = (const float*)d_in[20];
    const float* e_be = (const float*)d_in[21];

    const size_t P = (size_t)BB * CC * HWP;   // 16,777,216 floats per tensor
    const size_t Q = (size_t)BB * CC * 16;    // 65,536 floats (pooled / kernels)

    float* ws   = (float*)d_ws;
    float* yf   = ws;
    float* ye   = ws + P;
    float* gfb  = ws + 2 * P;
    float* geb  = ws + 3 * P;
    float* pf   = ws + 4 * P;
    float* pe   = pf + Q;
    float* kerf = pe + Q;
    float* kere = kerf + Q;
    float* syf  = kere + Q;        // 512 each: [sum | sumsq]
    float* sye  = syf + 512;
    float* szf  = sye + 512;
    float* sze  = szf + 512;
    float* abyf = sze + 512;       // 512 each: (a,b) pairs
    float* abye = abyf + 512;
    float* abzf = abye + 512;
    float* abze = abzf + 512;

    float* zf = (float*)d_out;     // bottleneck outputs land directly in d_out
    float* ze = zf + P;

    (void)hipMemsetAsync(syf, 0, 4 * 512 * sizeof(float), stream);

    // 1) adaptive pool to 4x4
    pool16<<<BB * CC, 256, 0, stream>>>(xf, pf);
    pool16<<<BB * CC, 256, 0, stream>>>(xe, pe);

    // 2) dynamic kernel generation (tiny WMMA GEMM, N=16)
    gemm256<16><<<dim3(1, 1, BB), 512, 0, stream>>>(pf, w_kf, b_kf, kerf, nullptr, 16);
    gemm256<16><<<dim3(1, 1, BB), 512, 0, stream>>>(pe, w_ke, b_ke, kere, nullptr, 16);

    // 3) redu 1x1 conv + BN stats (N tile = 32)
    gemm256<32><<<dim3(HWP / 32, 1, BB), 512, 0, stream>>>(xf, w_rf, b_rf, yf, syf, HWP);
    gemm256<32><<<dim3(HWP / 32, 1, BB), 512, 0, stream>>>(xe, w_re, b_re, ye, sye, HWP);

    const float inv_n = 1.0f / (float)((size_t)BB * HWP);
    bnfin<<<1, 256, 0, stream>>>(syf, g_rf, e_rf, abyf, inv_n);
    bnfin<<<1, 256, 0, stream>>>(sye, g_re, e_re, abye, inv_n);

    // 4) cross-guided dynamic filtering (BN+ReLU fused on load)
    dynconv<<<BB * CC, 256, 0, stream>>>(yf, abyf, kere, gfb);   // guide_f uses ker_e
    dynconv<<<BB * CC, 256, 0, stream>>>(ye, abye, kerf, geb);   // guide_e uses ker_f

    // 5) bottleneck 1x1 conv + BN stats, output straight into d_out
    gemm256<32><<<dim3(HWP / 32, 1, BB), 512, 0, stream>>>(gfb, w_bf, b_bf, zf, szf, HWP);
    gemm256<32><<<dim3(HWP / 32, 1, BB), 512, 0, stream>>>(geb, w_be, b_be, ze, sze, HWP);

    bnfin<<<1, 256, 0, stream>>>(szf, g_bf, e_bf, abzf, inv_n);
    bnfin<<<1, 256, 0, stream>>>(sze, g_be, e_be, abze, inv_n);

    // 6) final BN + ReLU in place
    bnrelu_ew<<<(unsigned)(P / 1024), 256, 0, stream>>>(zf, abzf);
    bnrelu_ew<<<(unsigned)(P / 1024), 256, 0, stream>>>(ze, abze);
}